// T_CIR_87471303950457
// MI455X (gfx1250) — compile-verified
//
#include <hip/hip_runtime.h>
#include <math.h>

typedef __attribute__((ext_vector_type(2))) float v2f;
typedef __attribute__((ext_vector_type(8))) float v8f;

#define Hh   64
#define Ww   64
#define Cc   64
#define Ff   64
#define KDIM 576      // 9 taps * 64 channels
#define LDSTR 580     // padded LDS stride: banks (4m + K) conflict-free, rows 16B aligned
#define PADMAX 65.0f  // H+1 clip bound (padded coords)

// ---------------------------------------------------------------- pass 0a: zero stats
__global__ void zero_stats_kernel(float* __restrict__ stats) {
    stats[threadIdx.x] = 0.0f;  // 256 floats: sum[2][64], sumsq[2][64]
}

// ---------------------------------------------------------------- pass 0b: transpose + pad weights
// okT[32][576]: okT[f*576+K] = (f<27) ? okern[K*27+f] : 0
// kT [64][576]: kT [f*576+K] = kern[K*64+f]
__global__ __launch_bounds__(256) void pack_weights_kernel(
    const float* __restrict__ okern, const float* __restrict__ kern,
    float* __restrict__ okT, float* __restrict__ kT)
{
    int idx = blockIdx.x * 256 + threadIdx.x;
    if (idx < 32 * KDIM) {
        int f = idx / KDIM, K = idx % KDIM;
        okT[idx] = (f < 27) ? okern[K * 27 + f] : 0.0f;
    } else if (idx < (32 + 64) * KDIM) {
        int j = idx - 32 * KDIM;
        int f = j / KDIM, K = j % KDIM;
        kT[j] = kern[K * Ff + f];
    }
}

// ---------------------------------------------------------------- pass 1: offset conv (3x3x64 -> 27) via fp32 WMMA
__global__ __launch_bounds__(64) void offset_conv_kernel(
    const float* __restrict__ x, const float* __restrict__ okT,
    const float* __restrict__ obias, float* __restrict__ offs)
{
    __shared__ float As[16 * LDSTR];

    const int bid = blockIdx.x;          // 2048 = 8 * 64 * 4
    const int n   = bid >> 8;
    const int h   = (bid >> 2) & 63;
    const int w0  = (bid & 3) << 4;
    const int tid = threadIdx.x;         // 64 threads
    const int c   = tid;                 // channel lane for staging

    // im2col stage: A[m=pixel][K=(tap*64+c)], zero-padded borders, coalesced over c
    for (int p = 0; p < 144; ++p) {
        int m = p / 9, t = p % 9;
        int ky = t / 3, kx = t % 3;
        int hh = h + ky - 1;
        int ww = w0 + m + kx - 1;
        float v = 0.0f;
        if (hh >= 0 && hh < Hh && ww >= 0 && ww < Ww)
            v = x[(((n * Hh + hh) * Ww + ww) * Cc) + c];
        As[m * LDSTR + t * Cc + c] = v;
    }
    __syncthreads();

    const int lane = tid & 31;
    const int wv   = tid >> 5;            // wave 0..1 -> output cols 0..15 / 16..31
    const int lm   = lane & 15;
    const int koff = (lane < 16) ? 0 : 2;
    const int f    = (wv << 4) + lm;
    const float* __restrict__ brow = okT + f * KDIM;   // transposed row, branch-free

    v8f acc = {};
#pragma unroll 4
    for (int kb = 0; kb < KDIM; kb += 4) {
        int K0 = kb + koff;
        v2f a = *(const v2f*)(&As[lm * LDSTR + K0]);   // ds_load_b64, 8B aligned
        v2f b = *(const v2f*)(brow + K0);              // global_load_b64
        acc = __builtin_amdgcn_wmma_f32_16x16x4_f32(false, a, false, b, (short)0,
                                                    acc, false, false);
    }

    if (f < 27) {
        float bval = obias[f];
        int mbase = (lane & 16) ? 8 : 0;
        for (int r = 0; r < 8; ++r) {
            int m = r + mbase;
            offs[(((n * Hh + h) * Ww + w0 + m) * 27) + f] = acc[r] + bval;
        }
    }
}

// ---------------------------------------------------------------- pass 2: bilinear sample + 576x64 GEMM via fp32 WMMA
__global__ __launch_bounds__(128) void dcn_gemm_kernel(
    const float* __restrict__ x, const float* __restrict__ offs,
    const float* __restrict__ kT, const float* __restrict__ bias,
    float* __restrict__ obuf, float* __restrict__ stats)
{
    __shared__ float As[16 * LDSTR];     // val[16 px][576]
    __shared__ float sW[16 * 9 * 4];     // mask-premultiplied bilinear weights
    __shared__ int   sA[16 * 9 * 4];     // flat gather base addresses (channel 0)

    const int bid = blockIdx.x;
    const int n   = bid >> 8;
    const int h   = (bid >> 2) & 63;
    const int w0  = (bid & 3) << 4;
    const int tid = threadIdx.x;         // 128 threads = 4 waves

    // ---- Step A: per (pixel, tap) corner weights & addresses
    for (int p = tid; p < 144; p += 128) {
        int m = p / 9, k = p % 9;
        int obase = ((n * Hh + h) * Ww + w0 + m) * 27;
        float dy  = offs[obase + 2 * k];
        float dx  = offs[obase + 2 * k + 1];
        float msk = 1.0f / (1.0f + __expf(-offs[obase + 18 + k]));
        int py = k / 3 - 1, px = k % 3 - 1;
        float gy = (float)(h + 1 + py) + dy;          // padded coords
        float gx = (float)(w0 + m + 1 + px) + dx;
        float fy = floorf(gy), fx = floorf(gx);
        float y0  = fminf(fmaxf(fy,        0.0f), PADMAX);
        float x0c = fminf(fmaxf(fx,        0.0f), PADMAX);
        float y1  = fminf(fmaxf(fy + 1.0f, 0.0f), PADMAX);
        float x1c = fminf(fmaxf(fx + 1.0f, 0.0f), PADMAX);
        float gyc = fminf(fmaxf(gy, 0.0f), PADMAX);
        float gxc = fminf(fmaxf(gx, 0.0f), PADMAX);
        float dy0 = gyc - y0, dy1 = y1 - gyc;
        float dx0 = gxc - x0c, dx1 = x1c - gxc;
        float wgt[4] = { dy0 * dx0, dy0 * dx1, dy1 * dx0, dy1 * dx1 };
        int cy[4] = { (int)y1, (int)y1, (int)y0, (int)y0 };
        int cx[4] = { (int)x1c, (int)x0c, (int)x1c, (int)x0c };
        int sbase = (m * 9 + k) * 4;
        for (int j = 0; j < 4; ++j) {
            int yy = cy[j] - 1, xx = cx[j] - 1;       // back to unpadded
            bool in = (yy >= 0 && yy < Hh && xx >= 0 && xx < Ww);
            sW[sbase + j] = in ? wgt[j] * msk : 0.0f; // padded-border sample == 0
            sA[sbase + j] = in ? ((n * Hh + yy) * Ww + xx) * Cc : 0;
        }
    }
    __syncthreads();

    // ---- Step B: build val matrix in LDS (float2 gather, coalesced over channels)
    {
        int c2 = (tid & 31) * 2;               // channel pair
        for (int p = (tid >> 5); p < 144; p += 4) {
            int m = p / 9, k = p % 9;
            int sbase = (m * 9 + k) * 4;
            float w0_ = sW[sbase + 0], w1_ = sW[sbase + 1];
            float w2_ = sW[sbase + 2], w3_ = sW[sbase + 3];
            v2f s0 = *(const v2f*)(x + sA[sbase + 0] + c2);
            v2f s1 = *(const v2f*)(x + sA[sbase + 1] + c2);
            v2f s2 = *(const v2f*)(x + sA[sbase + 2] + c2);
            v2f s3 = *(const v2f*)(x + sA[sbase + 3] + c2);
            v2f v;
            v.x = w0_ * s0.x + w1_ * s1.x + w2_ * s2.x + w3_ * s3.x;
            v.y = w0_ * s0.y + w1_ * s1.y + w2_ * s2.y + w3_ * s3.y;
            *(v2f*)(&As[m * LDSTR + k * Cc + c2]) = v;   // 8B aligned (stride even)
        }
    }
    __syncthreads();

    // ---- Step C: 16x64 = A[16x576] * kern[576x64], one 16-wide F tile per wave
    const int lane = tid & 31;
    const int wv   = tid >> 5;            // 0..3
    const int lm   = lane & 15;
    const int koff = (lane < 16) ? 0 : 2;
    const int f    = (wv << 4) + lm;
    const float* __restrict__ brow = kT + f * KDIM;    // transposed row

    v8f acc = {};
#pragma unroll 4
    for (int kb = 0; kb < KDIM; kb += 4) {
        int K0 = kb + koff;
        v2f a = *(const v2f*)(&As[lm * LDSTR + K0]);   // ds_load_b64
        v2f b = *(const v2f*)(brow + K0);              // global_load_b64
        acc = __builtin_amdgcn_wmma_f32_16x16x4_f32(false, a, false, b, (short)0,
                                                    acc, false, false);
    }

    // ---- Step D: bias, store pre-norm output, accumulate instance-norm stats
    float bval = bias[f];
    int mbase = (lane & 16) ? 8 : 0;
    float s = 0.0f, s2 = 0.0f;
    for (int r = 0; r < 8; ++r) {
        float v = acc[r] + bval;
        int m = r + mbase;
        obuf[(((n * Hh + h) * Ww + w0 + m) * Ff) + f] = v;
        s += v; s2 += v * v;
    }
    // lanes l and l+16 hold the same column f -> one cross-half-wave reduce
    s  += __shfl_xor(s, 16, 32);
    s2 += __shfl_xor(s2, 16, 32);
    if (lane < 16) {
        int bidx = n >> 2;                 // batch index (T=4)
        atomicAdd(&stats[bidx * Ff + f], s);
        atomicAdd(&stats[128 + bidx * Ff + f], s2);
    }
}

// ---------------------------------------------------------------- pass 3: instance norm + ReLU
__global__ __launch_bounds__(256) void norm_kernel(
    const float* __restrict__ obuf, const float* __restrict__ stats,
    const float* __restrict__ gamma, const float* __restrict__ beta,
    float* __restrict__ out)
{
    int idx = blockIdx.x * 256 + threadIdx.x;
    if (idx >= 2097152) return;            // 2*4*64*64*64
    int f = idx & 63;
    int b = idx >> 20;                     // T*H*W*F = 2^20
    const float inv_count = 1.0f / 16384.0f;   // T*H*W
    float mu  = stats[b * 64 + f] * inv_count;
    float var = stats[128 + b * 64 + f] * inv_count - mu * mu;
    float v = obuf[idx];
    float o = gamma[f] * (v - mu) * rsqrtf(var + 1e-3f) + beta[f];
    out[idx] = fmaxf(o, 0.0f);
}

// ---------------------------------------------------------------- launch
extern "C" void kernel_launch(void* const* d_in, const int* in_sizes, int n_in,
                              void* d_out, int out_size, void* d_ws, size_t ws_size,
                              hipStream_t stream) {
    const float* x     = (const float*)d_in[0];   // [2,4,64,64,64]
    const float* okern = (const float*)d_in[1];   // [3,3,64,27] -> [576,27]
    const float* obias = (const float*)d_in[2];   // [27]
    const float* kern  = (const float*)d_in[3];   // [3,3,64,64] -> [576,64]
    const float* bias  = (const float*)d_in[4];   // [64]
    const float* gamma = (const float*)d_in[5];   // [64]
    const float* beta  = (const float*)d_in[6];   // [64]
    float* out = (float*)d_out;

    float* offs  = (float*)d_ws;                  // 8*64*64*27  floats (3.5 MB)
    float* obuf  = offs + 8 * 64 * 64 * 27;       // 8*64*64*64  floats (8.4 MB)
    float* stats = obuf + 8 * 64 * 64 * 64;       // 256 floats
    float* okT   = stats + 256;                   // 32*576 floats (74 KB)
    float* kT    = okT + 32 * KDIM;               // 64*576 floats (147 KB)

    zero_stats_kernel<<<1, 256, 0, stream>>>(stats);
    pack_weights_kernel<<<(96 * KDIM + 255) / 256, 256, 0, stream>>>(okern, kern, okT, kT);
    offset_conv_kernel<<<2048, 64, 0, stream>>>(x, okT, obias, offs);
    dcn_gemm_kernel<<<2048, 128, 0, stream>>>(x, offs, kT, bias, obuf, stats);
    norm_kernel<<<8192, 256, 0, stream>>>(obuf, stats, gamma, beta, out);
}